// Convolution_59914793779562
// MI455X (gfx1250) — compile-verified
//
#include <hip/hip_runtime.h>
#include <hip/hip_bf16.h>

typedef float v2f __attribute__((ext_vector_type(2)));
typedef float v8f __attribute__((ext_vector_type(8)));

#define ISQ2 0.7071067811865476f
#define ISQ3 0.5773502691896258f
#define S6C  0.4082482904638631f
#define INV_SQRT_NEI 0.35355339059327373f   // 1/sqrt(8)
#define ISQ8 0.35355339059327373f
#define ISQ32 0.17677669529663687f
#define ISQ48 0.14433756729740643f

// ---------------------------------------------------------------------------
// Kernel 1: per-node self-interaction  s = node_s @ W1s/sqrt(32),
//           v = einsum('ncd,ck->nkd', node_v, W1v)/sqrt(16)
// Output feature layout F[n][80]: [0..31]=s, [32..79]=v (k*3+d)
// ---------------------------------------------------------------------------
__global__ __launch_bounds__(256) void node_kernel(
    const float* __restrict__ node_s, const float* __restrict__ node_v,
    const float* __restrict__ W1s, const float* __restrict__ W1v,
    float* __restrict__ F, int N)
{
    int n = blockIdx.x * blockDim.x + threadIdx.x;
    if (n >= N) return;
    float ns[32];
#pragma unroll
    for (int i = 0; i < 32; ++i) ns[i] = node_s[(long long)n * 32 + i];
    float* fo = F + (long long)n * 80;
    for (int o = 0; o < 32; ++o) {
        float a = 0.f;
#pragma unroll
        for (int i = 0; i < 32; ++i) a += ns[i] * W1s[i * 32 + o];
        fo[o] = a * ISQ32;
    }
    float nv[48];
#pragma unroll
    for (int i = 0; i < 48; ++i) nv[i] = node_v[(long long)n * 48 + i];
    for (int k = 0; k < 16; ++k) {
#pragma unroll
        for (int d = 0; d < 3; ++d) {
            float a = 0.f;
#pragma unroll
            for (int c = 0; c < 16; ++c) a += nv[c * 3 + d] * W1v[c * 16 + k];
            fo[32 + k * 3 + d] = a * 0.25f;
        }
    }
}

// ---------------------------------------------------------------------------
// WMMA helper: D += A[16xK] * B[KxN-tile], fp32 V_WMMA_F32_16X16X4_F32.
// A staged in LDS (row stride lda). B read lane-indexed from global weights.
// A layout (ISA 7.12.2, 32-bit A 16x4): lanes 0-15 hold K=k0,k0+1 in v0,v1;
// lanes 16-31 hold K=k0+2,k0+3.  B mirrors with N along lanes.
// ---------------------------------------------------------------------------
__device__ __forceinline__ v8f wmma_gemm(const float* A, int lda,
                                         const float* __restrict__ W, int ldw,
                                         int K, int n0, int lane, v8f c)
{
    const int row  = lane & 15;
    const int koff = (lane >> 4) << 1;     // 0 or 2
    const int n    = n0 + (lane & 15);
#pragma unroll
    for (int k0 = 0; k0 < K; k0 += 4) {
        v2f a, b;
        a.x = A[row * lda + k0 + koff];
        a.y = A[row * lda + k0 + koff + 1];
        b.x = W[(k0 + koff) * ldw + n];
        b.y = W[(k0 + koff + 1) * ldw + n];
        c = __builtin_amdgcn_wmma_f32_16x16x4_f32(false, a, false, b,
                                                  (short)0, c, false, false);
    }
    return c;
}

// silu via hardware reciprocal (v_rcp_f32) instead of IEEE division chain.
__device__ __forceinline__ float silu(float x) {
    return x * __builtin_amdgcn_rcpf(1.f + __expf(-x));
}

// ---------------------------------------------------------------------------
// Kernel 2: per-edge MLP (WMMA) + tensor products + atomic scatter-add.
// One wave = one 16-edge tile. 4 waves / block.
// ACC[n][320]: [0..15]=a0  [16..111]=a1o(c*3+d)  [112..159]=a1e
//              [160..239]=a2e(c*5+m)  [240..271]=as  [272..319]=av
// ---------------------------------------------------------------------------
#define EK_WAVES 4
#define EK_LDA   132   // padded row stride to dodge LDS bank conflicts

__global__ __launch_bounds__(128) void edge_kernel(
    const float* __restrict__ edge_vec, const float* __restrict__ edge_inv,
    const int* __restrict__ edge_src, const int* __restrict__ edge_dst,
    const float* __restrict__ Wm0, const float* __restrict__ Wm1,
    const float* __restrict__ Wm2, const float* __restrict__ F,
    float* __restrict__ ACC, int E)
{
    __shared__ float sbuf[EK_WAVES][16 * EK_LDA];
    const int lane = threadIdx.x & 31;
    const int wave = threadIdx.x >> 5;
    float* buf = sbuf[wave];
    const long long e0 = ((long long)blockIdx.x * EK_WAVES + wave) * 16;

    // ---- stage edge_inv tile [16 x 8] into LDS (A-matrix source) ----
#pragma unroll
    for (int idx = lane; idx < 16 * 8; idx += 32) {
        int e = idx >> 3, k = idx & 7;
        long long edge = e0 + e;
        buf[e * EK_LDA + k] = (edge < (long long)E) ? edge_inv[edge * 8 + k] : 0.f;
    }
    __syncthreads();

    const int colb   = lane & 15;
    const int rowoff = (lane >> 4) << 3;   // 0 or 8 (D-matrix row offset)

    // ---- layer 1: [16x8] @ [8x64], scale 1/sqrt(8), silu ----
    {
        v8f d[4] = {};
#pragma unroll
        for (int nt = 0; nt < 4; ++nt)
            d[nt] = wmma_gemm(buf, EK_LDA, Wm0, 64, 8, nt * 16, lane, d[nt]);
        __syncthreads();
#pragma unroll
        for (int nt = 0; nt < 4; ++nt)
#pragma unroll
            for (int j = 0; j < 8; ++j)
                buf[(j + rowoff) * EK_LDA + nt * 16 + colb] = silu(d[nt][j] * ISQ8);
    }
    __syncthreads();

    // ---- layer 2: [16x64] @ [64x64], scale 1/8, silu ----
    {
        v8f d[4] = {};
#pragma unroll
        for (int nt = 0; nt < 4; ++nt)
            d[nt] = wmma_gemm(buf, EK_LDA, Wm1, 64, 64, nt * 16, lane, d[nt]);
        __syncthreads();
#pragma unroll
        for (int nt = 0; nt < 4; ++nt)
#pragma unroll
            for (int j = 0; j < 8; ++j)
                buf[(j + rowoff) * EK_LDA + nt * 16 + colb] = silu(d[nt][j] * 0.125f);
    }
    __syncthreads();

    // ---- layer 3: [16x64] @ [64x128], scale 1/8, no activation -> f ----
    {
        v8f d[8] = {};
#pragma unroll
        for (int nt = 0; nt < 8; ++nt)
            d[nt] = wmma_gemm(buf, EK_LDA, Wm2, 128, 64, nt * 16, lane, d[nt]);
        __syncthreads();
#pragma unroll
        for (int nt = 0; nt < 8; ++nt)
#pragma unroll
            for (int j = 0; j < 8; ++j)
                buf[(j + rowoff) * EK_LDA + nt * 16 + colb] = d[nt][j] * 0.125f;
    }
    __syncthreads();

    // ---- tail: tensor products x factors, atomic scatter into ACC[dst] ----
    for (int e = 0; e < 16; ++e) {
        long long edge = e0 + e;
        if (edge >= (long long)E) break;          // wave-uniform
        const int src = edge_src[edge];
        const int dst = edge_dst[edge];
        const float u0 = edge_vec[edge * 3 + 0];
        const float u1 = edge_vec[edge * 3 + 1];
        const float u2 = edge_vec[edge * 3 + 2];
        const float* feat = F + (long long)src * 80;      // ss[0..31], vv at +32
        const float* f    = buf + e * EK_LDA;             // 128 factors
        float* acc        = ACC + (long long)dst * 320;

        for (int oi = lane; oi < 320; oi += 32) {
            float val;
            if (oi < 16) {                                 // 1o x 1o -> 0e
                int c = oi;
                float v0 = feat[32 + c * 3], v1 = feat[33 + c * 3], v2 = feat[34 + c * 3];
                val = ISQ3 * (v0 * u0 + v1 * u1 + v2 * u2) * f[c];
            } else if (oi < 112) {                         // 0e x 1o -> 1o
                int t = oi - 16, c = t / 3, dd = t - 3 * c;
                float uu = (dd == 0) ? u0 : ((dd == 1) ? u1 : u2);
                val = feat[c] * uu * f[16 + c];
            } else if (oi < 160) {                         // 1o x 1o -> 1e (cross)
                int t = oi - 112, c = t / 3, dd = t - 3 * c;
                float v0 = feat[32 + c * 3], v1 = feat[33 + c * 3], v2 = feat[34 + c * 3];
                float cr = (dd == 0) ? (v1 * u2 - v2 * u1)
                         : (dd == 1) ? (v2 * u0 - v0 * u2)
                                     : (v0 * u1 - v1 * u0);
                val = ISQ2 * cr * f[48 + c];
            } else if (oi < 240) {                         // 1o x 1o -> 2e
                int t = oi - 160, c = t / 5, m = t - 5 * c;
                float v0 = feat[32 + c * 3], v1 = feat[33 + c * 3], v2 = feat[34 + c * 3];
                float tp;
                if      (m == 0) tp = ISQ2 * (v0 * u1 + v1 * u0);
                else if (m == 1) tp = ISQ2 * (v1 * u2 + v2 * u1);
                else if (m == 2) tp = ISQ2 * (v0 * u2 + v2 * u0);
                else if (m == 3) tp = ISQ2 * (v0 * u0 - v1 * u1);
                else             tp = S6C * (2.f * v2 * u2 - v0 * u0 - v1 * u1);
                val = tp * f[64 + c];
            } else if (oi < 272) {                         // skip-scalar
                int c = oi - 240;
                val = feat[c] * f[80 + c];
            } else {                                       // skip-vector
                int t = oi - 272, c = t / 3, dd = t - 3 * c;
                val = feat[32 + c * 3 + dd] * f[96 + c];
            }
            atomicAdd(acc + oi, val);
        }
    }
}

// ---------------------------------------------------------------------------
// Kernel 3: per-node output linears. One block per node.
// ---------------------------------------------------------------------------
__global__ __launch_bounds__(256) void out_kernel(
    const float* __restrict__ ACC,
    const float* __restrict__ Wout_s, const float* __restrict__ Wout_1o,
    const float* __restrict__ Wout_1e, const float* __restrict__ Wout_2e,
    float* __restrict__ out, int N)
{
    __shared__ float L[320];
    const int n = blockIdx.x;
    for (int i = threadIdx.x; i < 320; i += 256)
        L[i] = ACC[(long long)n * 320 + i] * INV_SQRT_NEI;
    __syncthreads();

    const int o = threadIdx.x;
    if (o < 208) {
        float r = 0.f;
        if (o < 32) {                       // out_s = [a0 | as] @ Wout_s/sqrt48
            for (int c = 0; c < 48; ++c) {
                float in = (c < 16) ? L[c] : L[240 + (c - 16)];
                r += in * Wout_s[c * 32 + o];
            }
            r *= ISQ48;
        } else if (o < 80) {                // out_1o from [a1o | av]
            int t = o - 32, k = t / 3, d = t - 3 * k;
            for (int c = 0; c < 48; ++c) {
                float in = (c < 32) ? L[16 + c * 3 + d] : L[272 + (c - 32) * 3 + d];
                r += in * Wout_1o[c * 16 + k];
            }
            r *= ISQ48;
        } else if (o < 128) {               // out_1e
            int t = o - 80, k = t / 3, d = t - 3 * k;
            for (int c = 0; c < 16; ++c)
                r += L[112 + c * 3 + d] * Wout_1e[c * 16 + k];
            r *= 0.25f;
        } else {                            // out_2e
            int t = o - 128, k = t / 5, m = t - 5 * k;
            for (int c = 0; c < 16; ++c)
                r += L[160 + c * 5 + m] * Wout_2e[c * 16 + k];
            r *= 0.25f;
        }
        out[(long long)n * 208 + o] = r;
    }
}

// ---------------------------------------------------------------------------
extern "C" void kernel_launch(void* const* d_in, const int* in_sizes, int n_in,
                              void* d_out, int out_size, void* d_ws, size_t ws_size,
                              hipStream_t stream)
{
    const float* node_s   = (const float*)d_in[0];
    const float* node_v   = (const float*)d_in[1];
    const float* edge_vec = (const float*)d_in[2];
    const float* edge_inv = (const float*)d_in[3];
    const float* W1s      = (const float*)d_in[4];
    const float* W1v      = (const float*)d_in[5];
    const float* Wm0      = (const float*)d_in[6];
    const float* Wm1      = (const float*)d_in[7];
    const float* Wm2      = (const float*)d_in[8];
    const float* Wout_s   = (const float*)d_in[9];
    const float* Wout_1o  = (const float*)d_in[10];
    const float* Wout_1e  = (const float*)d_in[11];
    const float* Wout_2e  = (const float*)d_in[12];
    const int*   edge_src = (const int*)d_in[13];
    const int*   edge_dst = (const int*)d_in[14];

    const int N = in_sizes[0] / 32;
    const int E = in_sizes[13];

    float* F   = (float*)d_ws;                 // [N][80]   features
    float* ACC = F + (size_t)N * 80;           // [N][320]  accumulators

    (void)hipMemsetAsync(ACC, 0, (size_t)N * 320 * sizeof(float), stream);

    node_kernel<<<(N + 255) / 256, 256, 0, stream>>>(node_s, node_v, W1s, W1v, F, N);

    const int edgesPerBlock = EK_WAVES * 16;   // 64
    edge_kernel<<<(E + edgesPerBlock - 1) / edgesPerBlock, EK_WAVES * 32, 0, stream>>>(
        edge_vec, edge_inv, edge_src, edge_dst, Wm0, Wm1, Wm2, F, ACC, E);

    out_kernel<<<N, 256, 0, stream>>>(ACC, Wout_s, Wout_1o, Wout_1e, Wout_2e,
                                      (float*)d_out, N);
}